// BasicAttention_59639915872498
// MI455X (gfx1250) — compile-verified
//
#include <hip/hip_runtime.h>
#include <hip/hip_bf16.h>
#include <math.h>

// Problem dims (fixed by reference): B=64, N=2048, H=1024, A=512
#define BB 64
#define NN 2048
#define HH 1024
#define AA 512
#define NSPLIT 8

typedef float v2f __attribute__((ext_vector_type(2)));
typedef float v8f __attribute__((ext_vector_type(8)));

// ---------------------------------------------------------------------------
// gfx1250 hardware tanh (TRANS op). Builtin if available, else inline asm with
// a trailing v_nop to satisfy the TRANS-op result hazard (ISA 5 sec 7.4).
// ---------------------------------------------------------------------------
__device__ __forceinline__ float fast_tanh(float x) {
#if __has_builtin(__builtin_amdgcn_tanhf)
  return __builtin_amdgcn_tanhf(x);
#else
  float y;
  asm("v_tanh_f32 %0, %1\n\tv_nop" : "=v"(y) : "v"(x));
  return y;
#endif
}

// ---------------------------------------------------------------------------
// Kernel 1: w_h[b,a] = sum_h hs[b,h] * Wh[a,h]   (M=64, N=512, K=1024)
// One wave32 computes a 16x16 f32 tile via V_WMMA_F32_16X16X4_F32.
// A-frag layout (16x4 f32): lanes 0-15 hold M rows, lane-half selects K pair:
//   vgpr0/1 = K = 2*half, 2*half+1.  B-frag (4x16) mirrored with lanes = N.
// D layout: vgpr i -> M = i + 8*half, N = lane&15.
// ---------------------------------------------------------------------------
__global__ __launch_bounds__(256) void wh_gemm_wmma(
    const float* __restrict__ hs,   // [B,H]
    const float* __restrict__ Wh,   // [A,H]
    float* __restrict__ wh_out) {   // [B,A]
  const int wave = threadIdx.x >> 5;
  const int lane = threadIdx.x & 31;
  const int tile = blockIdx.x * 8 + wave;   // 128 tiles total
  const int mt = tile >> 5;                 // 0..3   (B/16)
  const int nt = tile & 31;                 // 0..31  (A/16)
  const int m0 = mt * 16, n0 = nt * 16;
  const int r = lane & 15;
  const int half = lane >> 4;

  const float* ap = hs + (size_t)(m0 + r) * HH + 2 * half;
  const float* bp = Wh + (size_t)(n0 + r) * HH + 2 * half;

  v8f c = {};
  for (int k = 0; k < HH; k += 4) {
    v2f a = *(const v2f*)(ap + k);
    v2f b = *(const v2f*)(bp + k);
    c = __builtin_amdgcn_wmma_f32_16x16x4_f32(
        /*neg_a=*/false, a, /*neg_b=*/false, b,
        /*c_mod=*/(short)0, c, /*reuse_a=*/false, /*reuse_b=*/false);
  }

  const int row = m0 + 8 * half;
  const int col = n0 + r;
#pragma unroll
  for (int i = 0; i < 8; ++i)
    wh_out[(size_t)(row + i) * AA + col] = c[i];
}

// ---------------------------------------------------------------------------
// Kernel 2: scores[b,n] = sum_a tanh(w_h[b,a] + p[b,n,a]) * Wa[a] + mask[b,n]
// One wave per (b,n) row; 8 rows per 256-thread block (same b).
// w_h[b,:] and W_alpha staged in LDS; p read once as float4 (memory-bound).
// ---------------------------------------------------------------------------
__global__ __launch_bounds__(256) void score_kernel(
    const float* __restrict__ p_att,   // [B,N,A]
    const float* __restrict__ mask,    // [B,N]
    const float* __restrict__ wh,      // [B,A]
    const float* __restrict__ Wa,      // [A]
    float* __restrict__ scores) {      // [B,N]
  __shared__ float s_wh[AA];
  __shared__ float s_wa[AA];
  const int b  = blockIdx.x >> 8;   // / (N/8)
  const int ng = blockIdx.x & 255;
  for (int i = threadIdx.x; i < AA; i += 256) {
    s_wh[i] = wh[(size_t)b * AA + i];
    s_wa[i] = Wa[i];
  }
  __syncthreads();

  const int wave = threadIdx.x >> 5;
  const int lane = threadIdx.x & 31;
  const int n = ng * 8 + wave;
  const float4* p4 = (const float4*)(p_att + ((size_t)b * NN + n) * AA);

  float sum = 0.f;
#pragma unroll
  for (int j = 0; j < 4; ++j) {
    const int a4 = j * 32 + lane;      // float4 index within the 512-row
    float4 v = p4[a4];
    const int a = a4 * 4;
    sum += fast_tanh(v.x + s_wh[a])     * s_wa[a];
    sum += fast_tanh(v.y + s_wh[a + 1]) * s_wa[a + 1];
    sum += fast_tanh(v.z + s_wh[a + 2]) * s_wa[a + 2];
    sum += fast_tanh(v.w + s_wh[a + 3]) * s_wa[a + 3];
  }
#pragma unroll
  for (int off = 16; off > 0; off >>= 1)
    sum += __shfl_xor(sum, off, 32);
  if (lane == 0)
    scores[(size_t)b * NN + n] = sum + mask[(size_t)b * NN + n];
}

// ---------------------------------------------------------------------------
// Kernel 3: in-place softmax over N per batch. One block per b.
// ---------------------------------------------------------------------------
__global__ __launch_bounds__(256) void softmax_kernel(float* __restrict__ scores) {
  __shared__ float red[8];
  const int b = blockIdx.x;
  float* row = scores + (size_t)b * NN;
  const int wave = threadIdx.x >> 5;
  const int lane = threadIdx.x & 31;

  float v[8];
  float m = -INFINITY;
#pragma unroll
  for (int j = 0; j < 8; ++j) {
    v[j] = row[threadIdx.x + j * 256];
    m = fmaxf(m, v[j]);
  }
#pragma unroll
  for (int off = 16; off > 0; off >>= 1)
    m = fmaxf(m, __shfl_xor(m, off, 32));
  if (lane == 0) red[wave] = m;
  __syncthreads();
  m = red[0];
#pragma unroll
  for (int w = 1; w < 8; ++w) m = fmaxf(m, red[w]);

  float s = 0.f;
#pragma unroll
  for (int j = 0; j < 8; ++j) {
    v[j] = __expf(v[j] - m);
    s += v[j];
  }
#pragma unroll
  for (int off = 16; off > 0; off >>= 1)
    s += __shfl_xor(s, off, 32);
  __syncthreads();
  if (lane == 0) red[wave] = s;
  __syncthreads();
  s = 0.f;
#pragma unroll
  for (int w = 0; w < 8; ++w) s += red[w];
  const float inv = 1.0f / s;
#pragma unroll
  for (int j = 0; j < 8; ++j)
    row[threadIdx.x + j * 256] = v[j] * inv;
}

// ---------------------------------------------------------------------------
// Kernel 4: partial[split][b][h] = sum_{n in split} attn[b,n]*att_feats[b,n,h]
// block = (b, split of 256 n's); thread owns 4 h's (float4) -> 1024 h covered.
// att_feats streamed once, fully coalesced (4 KB contiguous per n per block).
// Split-N partials keep the sum order deterministic (no float atomics).
// ---------------------------------------------------------------------------
__global__ __launch_bounds__(256) void att_partial(
    const float* __restrict__ att_feats,  // [B,N,H]
    const float* __restrict__ attn,       // [B,N]
    float* __restrict__ partial) {        // [NSPLIT,B,H]
  __shared__ float s_at[NN / NSPLIT];
  const int b = blockIdx.x >> 3;
  const int split = blockIdx.x & (NSPLIT - 1);
  const int n0 = split * (NN / NSPLIT);

  s_at[threadIdx.x] = attn[(size_t)b * NN + n0 + threadIdx.x];
  __syncthreads();

  const float4* p = (const float4*)(att_feats + ((size_t)b * NN + n0) * HH)
                    + threadIdx.x;  // h = 4*tid
  float4 acc = {0.f, 0.f, 0.f, 0.f};
#pragma unroll 4
  for (int n = 0; n < NN / NSPLIT; ++n) {
    const float w = s_at[n];
    const float4 x = p[(size_t)n * (HH / 4)];
    acc.x += w * x.x;
    acc.y += w * x.y;
    acc.z += w * x.z;
    acc.w += w * x.w;
  }
  float4* dst = (float4*)(partial + ((size_t)split * BB + b) * HH);
  dst[threadIdx.x] = acc;
}

// ---------------------------------------------------------------------------
// Kernel 5: out[b,h] = sum_split partial[split][b][h]
// ---------------------------------------------------------------------------
__global__ __launch_bounds__(256) void att_reduce(
    const float* __restrict__ partial, float* __restrict__ out) {
  const int idx = blockIdx.x * 256 + threadIdx.x;  // 0..B*H-1
  float s = 0.f;
#pragma unroll
  for (int j = 0; j < NSPLIT; ++j) s += partial[(size_t)j * (BB * HH) + idx];
  out[idx] = s;
}

// ---------------------------------------------------------------------------
extern "C" void kernel_launch(void* const* d_in, const int* in_sizes, int n_in,
                              void* d_out, int out_size, void* d_ws, size_t ws_size,
                              hipStream_t stream) {
  const float* hs        = (const float*)d_in[0];  // [B,H]
  const float* att_feats = (const float*)d_in[1];  // [B,N,H]
  const float* p_att     = (const float*)d_in[2];  // [B,N,A]
  const float* mask      = (const float*)d_in[3];  // [B,N]
  const float* Wh        = (const float*)d_in[4];  // [A,H]
  const float* Wa        = (const float*)d_in[5];  // [A]
  float* out = (float*)d_out;                      // [B,H]

  float* ws      = (float*)d_ws;
  float* wh      = ws;                           // B*A      = 32768 floats
  float* scores  = wh + (size_t)BB * AA;         // B*N      = 131072 floats
  float* partial = scores + (size_t)BB * NN;     // NSPLIT*B*H = 524288 floats

  wh_gemm_wmma<<<(BB / 16) * (AA / 16) / 8, 256, 0, stream>>>(hs, Wh, wh);
  score_kernel<<<BB * (NN / 8), 256, 0, stream>>>(p_att, mask, wh, Wa, scores);
  softmax_kernel<<<BB, 256, 0, stream>>>(scores);
  att_partial<<<BB * NSPLIT, 256, 0, stream>>>(att_feats, scores, partial);
  att_reduce<<<(BB * HH) / 256, 256, 0, stream>>>(partial, out);
}